// DehazeNet_30434138260000
// MI455X (gfx1250) — compile-verified
//
#include <hip/hip_runtime.h>
#include <math.h>

typedef __attribute__((ext_vector_type(2))) float v2f;
typedef __attribute__((ext_vector_type(8))) float v8f;

#define HSIZE 384
#define CHANNELS 3
#define HW2 (HSIZE*HSIZE)
#define CHW (CHANNELS*HW2)     // 442368
#define BATCH 2
#define TOTAL (BATCH*CHW)      // 884736

// ---------------------------------------------------------------------------
// Fused patch-attention stage.
//   tokens A[rows=4tok*9, K=D]  x  packed weights B[K=D, 3*heads*D]
//   -> qkv in LDS -> per-token softmax attention -> stage output image.
// WMMA: V_WMMA_F32_16X16X4_F32 (f32 in, f32 acc) to match reference precision.
// ---------------------------------------------------------------------------
template<int P, int HEADS, int D, int NUMH>
__global__ __launch_bounds__(32) void attn_stage(
    const float* __restrict__ x,
    const float* __restrict__ wq, const float* __restrict__ bq,
    const float* __restrict__ wk, const float* __restrict__ bk,
    const float* __restrict__ wv, const float* __restrict__ bv,
    const float* __restrict__ hw,
    float* __restrict__ out)
{
  constexpr int T     = 4;              // tokens per workgroup
  constexpr int ROWS  = T * 9;          // 36 GEMM rows
  constexpr int MT    = 3;              // M-tiles (rows padded to 48)
  constexpr int RP    = MT * 16;        // 48
  constexpr int KP    = (D + 3) & ~3;   // K padded to multiple of 4
  constexpr int KSTEPS= KP / 4;
  constexpr int HD    = HEADS * D;
  constexpr int COLS  = 3 * HD;         // q | k | v packed columns
  constexpr int NT    = (COLS + 15) / 16;
  constexpr int NP    = NT * 16;
  constexpr int PPQ   = P * P;

  __shared__ float Als[RP][KP];   // token tile
  __shared__ float Bls[KP][NP];   // packed Wq|Wk|Wv (transposed: [d][col])
  __shared__ float Qls[RP][NP];   // qkv result (+bias)

  const int lane = threadIdx.x;
  const int tok0 = blockIdx.x * T;

  // ---- pack weights: col = proj*HD + h*D + e ; Bls[d][col] = W[h][e][d] ----
  for (int i = lane; i < KP * NP; i += 32) {
    int d = i / NP, col = i - (i / NP) * NP;
    float val = 0.0f;
    if (d < D && col < COLS) {
      int proj = col / HD;
      int rem  = col - proj * HD;
      int h = rem / D, e = rem - (rem / D) * D;
      const float* W = (proj == 0) ? wq : (proj == 1) ? wk : wv;
      val = W[(h * D + e) * D + d];
    }
    Bls[d][col] = val;
  }

  // ---- gather tokens with reflect padding into A ----
  for (int i = lane; i < RP * KP; i += 32) {
    int row = i / KP, dd = i - (i / KP) * KP;
    float val = 0.0f;
    if (row < ROWS && dd < D) {
      int t  = tok0 + row / 9;
      int pp = row - (row / 9) * 9;
      int b  = t / (NUMH * NUMH);
      int rem= t - b * (NUMH * NUMH);
      int ti = rem / NUMH, tj = rem - (rem / NUMH) * NUMH;
      int c  = dd / PPQ;
      int r2 = dd - c * PPQ;
      int u  = r2 / P, v = r2 - (r2 / P) * P;
      int di = pp / 3, dj = pp - (pp / 3) * 3;
      int gy = (ti + di - 1) * P + u;
      int gx = (tj + dj - 1) * P + v;
      gy = gy < 0 ? -gy : (gy >= HSIZE ? 2 * HSIZE - 2 - gy : gy);
      gx = gx < 0 ? -gx : (gx >= HSIZE ? 2 * HSIZE - 2 - gx : gx);
      val = x[b * CHW + c * HW2 + gy * HSIZE + gx];
    }
    Als[row][dd] = val;
  }
  __syncthreads();

  // ---- WMMA GEMM: qkv = A x B  (all 32 lanes active; EXEC all-1s) ----
  const int mlane = lane & 15;
  const int half  = lane >> 4;

  for (int mt = 0; mt < MT; ++mt) {
    for (int nt = 0; nt < NT; ++nt) {
      v8f acc = {};
      #pragma unroll
      for (int ks = 0; ks < KSTEPS; ++ks) {
        // A 16x4 f32 layout: lane L -> m = L&15, element r -> k = (L>>4)*2 + r
        v2f af, bf;
        af.x = Als[mt * 16 + mlane][ks * 4 + half * 2 + 0];
        af.y = Als[mt * 16 + mlane][ks * 4 + half * 2 + 1];
        // B 4x16 f32 layout: lane L -> n = L&15, element r -> k = (L>>4)*2 + r
        bf.x = Bls[ks * 4 + half * 2 + 0][nt * 16 + mlane];
        bf.y = Bls[ks * 4 + half * 2 + 1][nt * 16 + mlane];
        acc = __builtin_amdgcn_wmma_f32_16x16x4_f32(
            false, af, false, bf, (short)0, acc, false, false);
      }
      // D 16x16 f32 layout: VGPR r -> row = (L>>4)*8 + r, col = L&15
      int col = nt * 16 + mlane;
      float bias = 0.0f;
      if (col < COLS) {
        int proj = col / HD;
        int rem  = col - proj * HD;
        int h = rem / D, e = rem - (rem / D) * D;
        const float* Bp = (proj == 0) ? bq : (proj == 1) ? bk : bv;
        bias = Bp[h * D + e];
      }
      #pragma unroll
      for (int r = 0; r < 8; ++r)
        Qls[mt * 16 + half * 8 + r][col] = acc[r] + bias;
    }
  }
  __syncthreads();

  // ---- per-token attention (one lane per token) ----
  if (lane < T) {
    int t  = tok0 + lane;
    int b  = t / (NUMH * NUMH);
    int rem= t - b * (NUMH * NUMH);
    int ti = rem / NUMH, tj = rem - (rem / NUMH) * NUMH;
    int rbase = lane * 9;
    const float scale = 1.0f / sqrtf((float)D);

    float mixed[D];
    #pragma unroll
    for (int e = 0; e < D; ++e) mixed[e] = 0.0f;

    for (int h = 0; h < HEADS; ++h) {
      float sc[9];
      float mx = -3.0e38f;
      #pragma unroll
      for (int p = 0; p < 9; ++p) {
        float s = 0.0f;
        for (int e = 0; e < D; ++e)
          s += Qls[rbase + 4][h * D + e] * Qls[rbase + p][HD + h * D + e];
        s *= scale;
        sc[p] = s;
        mx = fmaxf(mx, s);
      }
      float sum = 0.0f;
      #pragma unroll
      for (int p = 0; p < 9; ++p) { sc[p] = expf(sc[p] - mx); sum += sc[p]; }
      float wgt = hw[h] / sum;
      #pragma unroll
      for (int p = 0; p < 9; ++p) {
        float a = sc[p] * wgt;
        for (int e = 0; e < D; ++e)
          mixed[e] += a * Qls[rbase + p][2 * HD + h * D + e];
      }
    }
    for (int dd = 0; dd < D; ++dd) {
      int c  = dd / PPQ;
      int r2 = dd - c * PPQ;
      int u  = r2 / P, v = r2 - (r2 / P) * P;
      out[b * CHW + c * HW2 + (ti * P + u) * HSIZE + (tj * P + v)] = mixed[dd];
    }
  }
}

// ---------------------------------------------------------------------------
// conv0: cat([x9,x6,x3]) -> 5x5 conv pad 2 -> relu
// ---------------------------------------------------------------------------
__global__ void conv0_kernel(const float* __restrict__ x9, const float* __restrict__ x6,
                             const float* __restrict__ x3, const float* __restrict__ w,
                             const float* __restrict__ bias, float* __restrict__ x0)
{
  int idx = blockIdx.x * blockDim.x + threadIdx.x;
  if (idx >= TOTAL) return;
  int xx = idx % HSIZE;
  int t  = idx / HSIZE;
  int yy = t % HSIZE; t /= HSIZE;
  int oc = t % CHANNELS;
  int b  = t / CHANNELS;
  float acc = bias[oc];
  for (int ic = 0; ic < 9; ++ic) {
    const float* src = (ic < 3) ? x9 : (ic < 6) ? x6 : x3;
    int c = ic % 3;
    const float* sp = src + (b * CHANNELS + c) * HW2;
    const float* wp = w + (oc * 9 + ic) * 25;
    #pragma unroll
    for (int ky = 0; ky < 5; ++ky) {
      int iy = yy + ky - 2;
      if (iy < 0 || iy >= HSIZE) continue;
      #pragma unroll
      for (int kx = 0; kx < 5; ++kx) {
        int ix = xx + kx - 2;
        if (ix < 0 || ix >= HSIZE) continue;
        acc = fmaf(sp[iy * HSIZE + ix], wp[ky * 5 + kx], acc);
      }
    }
  }
  x0[idx] = fmaxf(acc, 0.0f);
}

// ---------------------------------------------------------------------------
// xg = per-batch max of x  (softmax over size-1 axis is identity)
// x >= 0 so integer-bit atomicMax on float bits is monotone.
// ---------------------------------------------------------------------------
__global__ void xg_init(int* xg) { if (threadIdx.x < BATCH) xg[threadIdx.x] = 0; }

__global__ void xg_reduce(const float* __restrict__ x, int* __restrict__ xg)
{
  int b = blockIdx.y;
  __shared__ float red[256];
  float m = 0.0f;
  for (int i = blockIdx.x * blockDim.x + threadIdx.x; i < CHW; i += gridDim.x * blockDim.x)
    m = fmaxf(m, x[b * CHW + i]);
  red[threadIdx.x] = m;
  __syncthreads();
  for (int s = 128; s > 0; s >>= 1) {
    if ((int)threadIdx.x < s) red[threadIdx.x] = fmaxf(red[threadIdx.x], red[threadIdx.x + s]);
    __syncthreads();
  }
  if (threadIdx.x == 0) atomicMax(&xg[b], __float_as_int(red[0]));
}

// ---------------------------------------------------------------------------
// out = relu(x * x0 + (xg - x0))
// ---------------------------------------------------------------------------
__global__ void final_kernel(const float* __restrict__ x, const float* __restrict__ x0,
                             const int* __restrict__ xg, float* __restrict__ out)
{
  int idx = blockIdx.x * blockDim.x + threadIdx.x;
  if (idx >= TOTAL) return;
  int b = idx / CHW;
  float g = __int_as_float(xg[b]);
  out[idx] = fmaxf(fmaf(x[idx], x0[idx], g - x0[idx]), 0.0f);
}

// ---------------------------------------------------------------------------
extern "C" void kernel_launch(void* const* d_in, const int* in_sizes, int n_in,
                              void* d_out, int out_size, void* d_ws, size_t ws_size,
                              hipStream_t stream)
{
  const float* x   = (const float*)d_in[0];
  const float* w3q = (const float*)d_in[1];  const float* b3q = (const float*)d_in[2];
  const float* w3k = (const float*)d_in[3];  const float* b3k = (const float*)d_in[4];
  const float* w3v = (const float*)d_in[5];  const float* b3v = (const float*)d_in[6];
  const float* hw3 = (const float*)d_in[7];
  const float* w6q = (const float*)d_in[8];  const float* b6q = (const float*)d_in[9];
  const float* w6k = (const float*)d_in[10]; const float* b6k = (const float*)d_in[11];
  const float* w6v = (const float*)d_in[12]; const float* b6v = (const float*)d_in[13];
  const float* hw6 = (const float*)d_in[14];
  const float* w9q = (const float*)d_in[15]; const float* b9q = (const float*)d_in[16];
  const float* w9k = (const float*)d_in[17]; const float* b9k = (const float*)d_in[18];
  const float* w9v = (const float*)d_in[19]; const float* b9v = (const float*)d_in[20];
  const float* hw9 = (const float*)d_in[21];
  const float* cw  = (const float*)d_in[22]; const float* cb  = (const float*)d_in[23];

  float* ws = (float*)d_ws;
  float* x3 = ws;
  float* x6 = ws + (size_t)TOTAL;
  float* x9 = ws + (size_t)2 * TOTAL;
  float* x0 = ws + (size_t)3 * TOTAL;
  int*   xg = (int*)(ws + (size_t)4 * TOTAL);

  attn_stage<1, 6, 3, 384><<<(BATCH * 384 * 384) / 4, 32, 0, stream>>>(
      x, w3q, b3q, w3k, b3k, w3v, b3v, hw3, x3);
  attn_stage<2, 4, 12, 192><<<(BATCH * 192 * 192) / 4, 32, 0, stream>>>(
      x3, w6q, b6q, w6k, b6k, w6v, b6v, hw6, x6);
  attn_stage<3, 2, 27, 128><<<(BATCH * 128 * 128) / 4, 32, 0, stream>>>(
      x6, w9q, b9q, w9k, b9k, w9v, b9v, hw9, x9);

  conv0_kernel<<<(TOTAL + 255) / 256, 256, 0, stream>>>(x9, x6, x3, cw, cb, x0);

  xg_init<<<1, 32, 0, stream>>>(xg);
  dim3 rg(64, BATCH);
  xg_reduce<<<rg, 256, 0, stream>>>(x, xg);

  final_kernel<<<(TOTAL + 255) / 256, 256, 0, stream>>>(x, x0, xg, (float*)d_out);
}